// GConvBlock_68547678044321
// MI455X (gfx1250) — compile-verified
//
#include <hip/hip_runtime.h>
#include <hip/hip_bf16.h>
#include <math.h>

// Problem constants (from reference)
#define BB   8
#define NN   1024
#define LL   4
#define CC   128
#define GG   128
#define OO   64
#define FF   64
#define ALPHA 0.2f
#define EPSN  1e-5f
#define NEGV  (-9e15f)

typedef __attribute__((ext_vector_type(2))) float v2f;
typedef __attribute__((ext_vector_type(8))) float v8f;

__device__ __forceinline__ float lrelu(float x) { return x > 0.f ? x : ALPHA * x; }

// CDNA5 async global->LDS copy (ASYNCcnt-tracked, bypasses VGPRs).
// lds_addr: 32-bit LDS byte offset (flat shared pointer truncated per ISA 10.2)
__device__ __forceinline__ void async_load_b128(unsigned lds_addr, const void* gptr) {
    asm volatile("global_load_async_to_lds_b128 %0, %1, off"
                 :: "v"(lds_addr), "v"((unsigned long long)(uintptr_t)gptr)
                 : "memory");
}
__device__ __forceinline__ void wait_async0() {
    asm volatile("s_wait_asynccnt 0x0" ::: "memory");
}

// ---------------------------------------------------------------------------
// K1: gs = global_state @ Wg ; g2[l,b,f] = gs[b] @ W[l,128:,:] ; bn_global+relu
// one block, 512 threads
// ---------------------------------------------------------------------------
__global__ void k_global(const float* __restrict__ gstate, const float* __restrict__ Wg,
                         const float* __restrict__ W, const float* __restrict__ gamma_g,
                         const float* __restrict__ beta_g, float* __restrict__ ws_gs,
                         float* __restrict__ ws_g2, float* __restrict__ d_out) {
    __shared__ float s_gs[BB * OO];
    int tid = threadIdx.x;
    int b = tid >> 6, f = tid & 63;
    float acc = 0.f;
    #pragma unroll 8
    for (int o = 0; o < GG; ++o) acc += gstate[b * GG + o] * Wg[o * OO + f];
    s_gs[b * OO + f] = acc;
    ws_gs[b * OO + f] = acc;
    __syncthreads();
    // g2[l,b,f] = sum_o gs[b,o] * W[l, 128+o, f]
    for (int idx = tid; idx < LL * BB * OO; idx += 512) {
        int l = idx >> 9, b2 = (idx >> 6) & 7, f2 = idx & 63;
        float a = 0.f;
        #pragma unroll 8
        for (int o = 0; o < OO; ++o)
            a += s_gs[b2 * OO + o] * W[(l * (CC + OO) + CC + o) * FF + f2];
        ws_g2[idx] = a;
    }
    // bn_global over batch (8 samples per feature) + relu
    if (tid < OO) {
        float mu = 0.f;
        #pragma unroll
        for (int bb = 0; bb < BB; ++bb) mu += s_gs[bb * OO + tid];
        mu *= (1.f / BB);
        float var = 0.f;
        #pragma unroll
        for (int bb = 0; bb < BB; ++bb) {
            float d = s_gs[bb * OO + tid] - mu;
            var += d * d;
        }
        var *= (1.f / BB);
        float sc = gamma_g[tid] * rsqrtf(var + EPSN);
        #pragma unroll
        for (int bb = 0; bb < BB; ++bb) {
            float y = (s_gs[bb * OO + tid] - mu) * sc + beta_g[tid];
            d_out[BB * NN * FF + bb * OO + tid] = fmaxf(y, 0.f);
        }
    }
}

// ---------------------------------------------------------------------------
// K2: h[l,b,n,f] = V[b,n,:] @ W[l,:128,f] + (n<gsz[b])*g2[l,b,f]
// WMMA f32 16x16x4. grid (N/128, B, L), block 256 (8 waves, 16 rows each)
// ---------------------------------------------------------------------------
__global__ void k_h(const float* __restrict__ V, const float* __restrict__ W,
                    const float* __restrict__ g2, const int* __restrict__ gsz,
                    float* __restrict__ h) {
    __shared__ float sW[CC * FF];   // 32 KB
    __shared__ float sg2[FF];
    __shared__ int s_gsz;
    int tid = threadIdx.x;
    int b = blockIdx.y, l = blockIdx.z;
    for (int j = tid; j < CC * FF; j += 256) sW[j] = W[l * (CC + OO) * FF + j];
    if (tid < FF) sg2[tid] = g2[(l * BB + b) * OO + tid];
    if (tid == 0) s_gsz = gsz[b];
    __syncthreads();

    int w = tid >> 5, lane = tid & 31;
    int row = lane & 15, half = lane >> 4;
    int n0 = blockIdx.x * 128 + w * 16;
    const float* vrow = V + (size_t)(b * NN + n0 + row) * CC;

    v8f acc[4] = {};
    for (int k = 0; k < CC; k += 4) {
        v2f a;
        a.x = vrow[k + half * 2];
        a.y = vrow[k + half * 2 + 1];
        #pragma unroll
        for (int ft = 0; ft < 4; ++ft) {
            v2f bb;
            bb.x = sW[(k + half) * FF + ft * 16 + row];
            bb.y = sW[(k + 2 + half) * FF + ft * 16 + row];
            acc[ft] = __builtin_amdgcn_wmma_f32_16x16x4_f32(
                false, a, false, bb, (short)0, acc[ft], false, false);
        }
    }
    // epilogue: add masked global-feature contribution, store h
    #pragma unroll
    for (int ft = 0; ft < 4; ++ft) {
        #pragma unroll
        for (int r = 0; r < 8; ++r) {
            int rr = r + half * 8;
            int gn = n0 + rr;
            float val = acc[ft][r];
            if (gn < s_gsz) val += sg2[ft * 16 + row];
            h[((size_t)(l * BB + b) * NN + gn) * FF + ft * 16 + row] = val;
        }
    }
}

// ---------------------------------------------------------------------------
// K3: f1[l,b,n] = h[l,b,n,:]·a1[l,:] ; f2 likewise. 1 wave per row.
// ---------------------------------------------------------------------------
__global__ void k_f12(const float* __restrict__ h, const float* __restrict__ a1,
                      const float* __restrict__ a2, float* __restrict__ f1,
                      float* __restrict__ f2) {
    int tid = threadIdx.x;
    int w = tid >> 5, lane = tid & 31;
    int idx = blockIdx.x * 8 + w;          // row in [0, L*B*N)
    int l = idx >> 13;                     // / (B*N)
    const float* hr = h + (size_t)idx * FF;
    float h0 = hr[lane], h1 = hr[32 + lane];
    float s1 = h0 * a1[l * FF + lane] + h1 * a1[l * FF + 32 + lane];
    float s2 = h0 * a2[l * FF + lane] + h1 * a2[l * FF + 32 + lane];
    #pragma unroll
    for (int off = 16; off > 0; off >>= 1) {
        s1 += __shfl_xor(s1, off, 32);
        s2 += __shfl_xor(s2, off, 32);
    }
    if (lane == 0) { f1[idx] = s1; f2[idx] = s2; }
}

// ---------------------------------------------------------------------------
// K4a: partial online column softmax stats over a 256-row n-chunk
// grid (N/256, B, L*4); blockIdx.z = l*4 + chunk. Coalesced pass over A.
// ---------------------------------------------------------------------------
__global__ void k_colstats(const float* __restrict__ A, const float* __restrict__ f1,
                           const float* __restrict__ f2, float* __restrict__ pm,
                           float* __restrict__ ps) {
    __shared__ float sf1[256];
    int tid = threadIdx.x;
    int b = blockIdx.y;
    int l = blockIdx.z >> 2, ch = blockIdx.z & 3;
    int m = blockIdx.x * 256 + tid;
    int nbase = ch * 256;
    float f2v = f2[(l * BB + b) * NN + m];
    sf1[tid] = f1[(l * BB + b) * NN + nbase + tid];
    __syncthreads();
    float mx = -__builtin_inff(), sum = 0.f;
    #pragma unroll 4
    for (int j = 0; j < 256; ++j) {
        int n = nbase + j;
        float av = A[(((size_t)b * NN + n) * LL + l) * NN + m];
        float e = lrelu(sf1[j] + f2v);
        float val = av > 0.f ? e : NEGV;
        float nm = fmaxf(mx, val);
        sum = sum * __expf(mx - nm) + __expf(val - nm);
        mx = nm;
    }
    int ci = ((ch * LL + l) * BB + b) * NN + m;
    pm[ci] = mx;
    ps[ci] = sum;
}

// K4b: merge the 4 partials per column: m = max(m_c), s = sum s_c*exp(m_c-m)
__global__ void k_colmerge(const float* __restrict__ pm, const float* __restrict__ ps,
                           float* __restrict__ cm, float* __restrict__ cs) {
    int idx = blockIdx.x * 256 + threadIdx.x;   // (l*BB+b)*NN+m, 32768 total
    const int S = LL * BB * NN;
    float mx = -__builtin_inff();
    #pragma unroll
    for (int c = 0; c < 4; ++c) mx = fmaxf(mx, pm[c * S + idx]);
    float s = 0.f;
    #pragma unroll
    for (int c = 0; c < 4; ++c) s += ps[c * S + idx] * __expf(pm[c * S + idx] - mx);
    cm[idx] = mx;
    cs[idx] = s;
}

// ---------------------------------------------------------------------------
// K5: out[b,n,f] = sum_l sum_m att[l,b,n,m] * h[l,b,m,f] + bias_gat[f]
// WMMA f32 16x16x4; att recomputed on the fly; A rows + h tile brought into
// LDS with GLOBAL_LOAD_ASYNC_TO_LDS_B128 (ASYNCcnt), freeing VALU/VGPRs.
// grid (N/128, B), block 256 (8 waves, 16 output rows each)
// ---------------------------------------------------------------------------
__global__ void k_attn(const float* __restrict__ A, const float* __restrict__ h,
                       const float* __restrict__ f1, const float* __restrict__ f2,
                       const float* __restrict__ cm, const float* __restrict__ cs,
                       const float* __restrict__ bias, float* __restrict__ tmp) {
    __shared__ float sh[64 * FF];        // 16 KB  h tile [64 m][64 f]
    __shared__ float sA[8][16 * 64];     // 32 KB  per-wave A rows [16 n][64 m]
    __shared__ float scm[64], scs[64], sf2[64];
    int tid = threadIdx.x;
    int w = tid >> 5, lane = tid & 31;
    int row = lane & 15, half = lane >> 4;
    int b = blockIdx.y;
    int n0 = blockIdx.x * 128 + w * 16;

    v8f acc[4] = {};
    for (int l = 0; l < LL; ++l) {
        float f1v = f1[(l * BB + b) * NN + n0 + row];
        for (int mt = 0; mt < NN / 64; ++mt) {
            __syncthreads();
            // h tile: 4096 floats, async direct-to-LDS (4x b128 per thread)
            const float* hp = h + ((size_t)(l * BB + b) * NN + mt * 64) * FF;
            #pragma unroll
            for (int j = 0; j < 4; ++j) {
                int e = (tid + j * 256) * 4;
                async_load_b128((unsigned)(uintptr_t)&sh[e], hp + e);
            }
            // per-wave A rows: 16 x 64 floats; each iteration covers 2 rows
            const float* ap = A + (((size_t)(b * NN + n0) * LL + l) * NN + mt * 64);
            #pragma unroll
            for (int j = lane; j < 256; j += 32) {
                int r = j >> 4, mm4 = (j & 15) * 4;
                async_load_b128((unsigned)(uintptr_t)&sA[w][r * 64 + mm4],
                                ap + (size_t)r * LL * NN + mm4);
            }
            if (tid < 64) {
                int mi = (l * BB + b) * NN + mt * 64 + tid;
                scm[tid] = cm[mi];
                scs[tid] = cs[mi];
                sf2[tid] = f2[mi];
            }
            wait_async0();          // barriers do not wait on ASYNCcnt
            __syncthreads();
            for (int km = 0; km < 64; km += 4) {
                int m1 = km + half * 2, m2 = m1 + 1;
                float e1 = lrelu(f1v + sf2[m1]);
                float e2 = lrelu(f1v + sf2[m2]);
                float v1 = sA[w][row * 64 + m1] > 0.f ? e1 : NEGV;
                float v2 = sA[w][row * 64 + m2] > 0.f ? e2 : NEGV;
                v2f a;
                a.x = __expf(v1 - scm[m1]) / scs[m1];
                a.y = __expf(v2 - scm[m2]) / scs[m2];
                #pragma unroll
                for (int ft = 0; ft < 4; ++ft) {
                    v2f bb;
                    bb.x = sh[(km + half) * FF + ft * 16 + row];
                    bb.y = sh[(km + 2 + half) * FF + ft * 16 + row];
                    acc[ft] = __builtin_amdgcn_wmma_f32_16x16x4_f32(
                        false, a, false, bb, (short)0, acc[ft], false, false);
                }
            }
        }
    }
    #pragma unroll
    for (int ft = 0; ft < 4; ++ft) {
        #pragma unroll
        for (int r = 0; r < 8; ++r) {
            int rr = r + half * 8;
            int f = ft * 16 + row;
            tmp[((size_t)b * NN + n0 + rr) * FF + f] = acc[ft][r] + bias[f];
        }
    }
}

// ---------------------------------------------------------------------------
// K6: per-feature sum / sumsq of tmp via LDS + global atomics
// ---------------------------------------------------------------------------
__global__ void k_stats(const float* __restrict__ tmp, float* __restrict__ gsum,
                        float* __restrict__ gsq) {
    __shared__ float ssum[FF], ssq[FF];
    int tid = threadIdx.x;
    if (tid < FF) { ssum[tid] = 0.f; ssq[tid] = 0.f; }
    __syncthreads();
    const int total = BB * NN * FF;
    for (int idx = blockIdx.x * 256 + tid; idx < total; idx += 256 * 256) {
        float v = tmp[idx];
        int f = idx & 63;
        atomicAdd(&ssum[f], v);
        atomicAdd(&ssq[f], v * v);
    }
    __syncthreads();
    if (tid < FF) {
        atomicAdd(&gsum[tid], ssum[tid]);
        atomicAdd(&gsq[tid], ssq[tid]);
    }
}

// ---------------------------------------------------------------------------
// K7: BN (batch stats) + relu -> d_out
// ---------------------------------------------------------------------------
__global__ void k_bn(const float* __restrict__ tmp, const float* __restrict__ gsum,
                     const float* __restrict__ gsq, const float* __restrict__ gamma_n,
                     const float* __restrict__ beta_n, float* __restrict__ d_out) {
    int idx = blockIdx.x * 256 + threadIdx.x;
    int f = idx & 63;
    const float inv = 1.f / (float)(BB * NN);
    float mu = gsum[f] * inv;
    float var = gsq[f] * inv - mu * mu;
    float y = (tmp[idx] - mu) * rsqrtf(var + EPSN) * gamma_n[f] + beta_n[f];
    d_out[idx] = fmaxf(y, 0.f);
}

// ---------------------------------------------------------------------------
extern "C" void kernel_launch(void* const* d_in, const int* in_sizes, int n_in,
                              void* d_out, int out_size, void* d_ws, size_t ws_size,
                              hipStream_t stream) {
    const float* V      = (const float*)d_in[0];
    const float* A      = (const float*)d_in[1];
    const float* gstate = (const float*)d_in[2];
    const int*   gsz    = (const int*)d_in[3];
    // d_in[4] = subgraph_size (unused)
    const float* Wg     = (const float*)d_in[5];
    const float* W      = (const float*)d_in[6];
    const float* a1     = (const float*)d_in[7];
    const float* a2     = (const float*)d_in[8];
    const float* bias   = (const float*)d_in[9];
    const float* gamma_n = (const float*)d_in[10];
    const float* beta_n  = (const float*)d_in[11];
    const float* gamma_g = (const float*)d_in[12];
    const float* beta_g  = (const float*)d_in[13];
    float* out = (float*)d_out;

    float* ws = (float*)d_ws;
    float* ws_gs  = ws;                                   // 512
    float* ws_g2  = ws_gs + 512;                          // 2048
    float* ws_h   = ws + 4096;                            // L*B*N*F = 2,097,152
    float* ws_f1  = ws_h + (size_t)LL * BB * NN * FF;     // 32768
    float* ws_f2  = ws_f1 + LL * BB * NN;                 // 32768
    float* ws_cm  = ws_f2 + LL * BB * NN;                 // 32768
    float* ws_cs  = ws_cm + LL * BB * NN;                 // 32768
    float* ws_tmp = ws_cs + LL * BB * NN;                 // B*N*F = 524288
    float* ws_st  = ws_tmp + (size_t)BB * NN * FF;        // 128 (sum|sumsq)
    float* ws_pm  = ws_st + 128;                          // 4*L*B*N = 131072
    float* ws_ps  = ws_pm + 4 * LL * BB * NN;             // 131072

    hipLaunchKernelGGL(k_global, dim3(1), dim3(512), 0, stream,
                       gstate, Wg, W, gamma_g, beta_g, ws_gs, ws_g2, out);
    hipLaunchKernelGGL(k_h, dim3(NN / 128, BB, LL), dim3(256), 0, stream,
                       V, W, ws_g2, gsz, ws_h);
    hipLaunchKernelGGL(k_f12, dim3(LL * BB * NN / 8), dim3(256), 0, stream,
                       ws_h, a1, a2, ws_f1, ws_f2);
    hipLaunchKernelGGL(k_colstats, dim3(NN / 256, BB, LL * 4), dim3(256), 0, stream,
                       A, ws_f1, ws_f2, ws_pm, ws_ps);
    hipLaunchKernelGGL(k_colmerge, dim3(LL * BB * NN / 256), dim3(256), 0, stream,
                       ws_pm, ws_ps, ws_cm, ws_cs);
    hipLaunchKernelGGL(k_attn, dim3(NN / 128, BB), dim3(256), 0, stream,
                       A, ws_h, ws_f1, ws_f2, ws_cm, ws_cs, bias, ws_tmp);
    hipMemsetAsync(ws_st, 0, 2 * FF * sizeof(float), stream);
    hipLaunchKernelGGL(k_stats, dim3(256), dim3(256), 0, stream,
                       ws_tmp, ws_st, ws_st + FF);
    hipLaunchKernelGGL(k_bn, dim3(BB * NN * FF / 256), dim3(256), 0, stream,
                       ws_tmp, ws_st, ws_st + FF, gamma_n, beta_n, out);
}